// SportsGNN_2001454760227
// MI455X (gfx1250) — compile-verified
//
#include <hip/hip_runtime.h>
#include <math.h>

typedef float v2f __attribute__((ext_vector_type(2)));
typedef float v8f __attribute__((ext_vector_type(8)));

#define LRELU_MLP(v) ((v) > 0.f ? (v) : 0.1f * (v))
#define LRELU_GAT(v) ((v) > 0.f ? (v) : 0.2f * (v))

// Sign-aware float atomic max via integer atomics (valid with -inf init):
//  - nonneg values: signed-int max (neg floats have sign bit -> negative ints)
//  - negative values: unsigned min (more-negative float -> larger uint)
__device__ __forceinline__ void atomicMaxF(float* addr, float val) {
  if (val >= 0.f)
    atomicMax((int*)addr, __float_as_int(val));
  else
    atomicMin((unsigned int*)addr, __float_as_uint(val));
}

// ---------------------------------------------------------------------------
// Node feature MLP: x[N,3] -> h[N,8], stored zero-padded to 16 columns so the
// WMMA K-sweep (K=8, then K=16) is safe for both GAT layers.
// ---------------------------------------------------------------------------
__global__ void k_node(const float* __restrict__ x,
                       const float* __restrict__ w1, const float* __restrict__ b1,
                       const float* __restrict__ w2, const float* __restrict__ b2,
                       float* __restrict__ h, int N, int Npad) {
  int i = blockIdx.x * blockDim.x + threadIdx.x;
  if (i >= Npad) return;
  if (i >= N) {  // pad rows so full 16-row WMMA tiles read zeros
    for (int j = 0; j < 16; ++j) h[(size_t)i * 16 + j] = 0.f;
    return;
  }
  float x0 = x[i * 3 + 0], x1 = x[i * 3 + 1], x2 = x[i * 3 + 2];
  float t[8];
#pragma unroll
  for (int j = 0; j < 8; ++j) {
    float v = w1[j * 3 + 0] * x0 + w1[j * 3 + 1] * x1 + w1[j * 3 + 2] * x2 + b1[j];
    t[j] = LRELU_MLP(v);
  }
#pragma unroll
  for (int j = 0; j < 8; ++j) {
    float v = b2[j];
#pragma unroll
    for (int k = 0; k < 8; ++k) v += w2[j * 8 + k] * t[k];
    h[(size_t)i * 16 + j] = LRELU_MLP(v);
    h[(size_t)i * 16 + 8 + j] = 0.f;
  }
}

// ---------------------------------------------------------------------------
// Dense lifts via V_WMMA_F32_16X16X4_F32: one wave = one 16-node tile.
// xl = h @ wl.T + bl ; xr = h @ wr.T + br   (shared A operand, 2 WMMAs/K-step)
// A layout (16x4 f32): row M = lane&15; VGPR0 holds K=(lane<16?0:2)+kb,
// VGPR1 holds K+1.  B layout (4x16): col N = lane&15, same K striping.
// C/D: col N = lane&15, row M = r + 8*(lane>=16).
// ---------------------------------------------------------------------------
__global__ void k_lin(const float* __restrict__ h,
                      const float* __restrict__ wl, const float* __restrict__ bl,
                      const float* __restrict__ wr, const float* __restrict__ br,
                      float* __restrict__ xl, float* __restrict__ xr,
                      int kdim, int ntiles) {
  int wave = (int)((blockIdx.x * blockDim.x + threadIdx.x) >> 5);
  int lane = threadIdx.x & 31;
  if (wave >= ntiles) return;  // wave-uniform guard: EXEC stays all-ones
  int n = lane & 15;           // A row index == B/C column index for this lane
  int hi = lane >> 4;
  const float* hrow = h + (size_t)(wave * 16 + n) * 16;
  v8f cl, cr;
  float bln = bl[n], brn = br[n];
#pragma unroll
  for (int r = 0; r < 8; ++r) { cl[r] = bln; cr[r] = brn; }
  for (int kb = 0; kb < kdim; kb += 4) {  // kdim is 8 or 16
    int k0 = kb + hi * 2;
    v2f a, bbl, bbr;
    a[0] = hrow[k0];
    a[1] = hrow[k0 + 1];
    bbl[0] = wl[n * kdim + k0];
    bbl[1] = wl[n * kdim + k0 + 1];
    bbr[0] = wr[n * kdim + k0];
    bbr[1] = wr[n * kdim + k0 + 1];
    cl = __builtin_amdgcn_wmma_f32_16x16x4_f32(false, a, false, bbl, (short)0, cl,
                                               false, false);
    cr = __builtin_amdgcn_wmma_f32_16x16x4_f32(false, a, false, bbr, (short)0, cr,
                                               false, false);
  }
#pragma unroll
  for (int r = 0; r < 8; ++r) {
    int m = r + hi * 8;
    size_t o = (size_t)(wave * 16 + m) * 16 + n;
    xl[o] = cl[r];
    xr[o] = cr[r];
  }
}

// ---------------------------------------------------------------------------
// Per-layer init: segment-max to -inf, denom/accum to 0; optionally zero pool.
// ---------------------------------------------------------------------------
__global__ void k_init(float* __restrict__ nmax, float* __restrict__ den,
                       float* __restrict__ acc, float* __restrict__ pool, int N) {
  int i = blockIdx.x * blockDim.x + threadIdx.x;
  if (pool != nullptr && i < 16) pool[i] = 0.f;
  if (i >= N) return;
  float ninf = -__builtin_inff();
  nmax[i * 2 + 0] = ninf;
  nmax[i * 2 + 1] = ninf;
  den[i * 2 + 0] = 0.f;
  den[i * 2 + 1] = 0.f;
#pragma unroll
  for (int j = 0; j < 16; ++j) acc[(size_t)i * 16 + j] = 0.f;
}

// ---------------------------------------------------------------------------
// Edge pass 1: logits[e,h] = sum_o lrelu0.2(xl[src]+xr[dst]+We*e)[h,o]*att[h,o]
// store logits; segment max into nmax[dst,h].
// ---------------------------------------------------------------------------
__global__ void k_e1(const int* __restrict__ ei, const float* __restrict__ ea,
                     const float* __restrict__ xl, const float* __restrict__ xr,
                     const float* __restrict__ we, const float* __restrict__ att,
                     float* __restrict__ elog, float* __restrict__ nmax, int E) {
  int e = blockIdx.x * blockDim.x + threadIdx.x;
  if (e >= E) return;
  int s = ei[e], d = ei[E + e];
  float e0 = ea[(size_t)e * 2 + 0], e1 = ea[(size_t)e * 2 + 1];
  const float* xls = xl + (size_t)s * 16;
  const float* xrd = xr + (size_t)d * 16;
#pragma unroll
  for (int hh = 0; hh < 2; ++hh) {
    float lg = 0.f;
#pragma unroll
    for (int o = 0; o < 8; ++o) {
      int j = hh * 8 + o;
      float v = xls[j] + xrd[j] + we[j * 2 + 0] * e0 + we[j * 2 + 1] * e1;
      v = LRELU_GAT(v);
      lg += v * att[j];
    }
    elog[(size_t)e * 2 + hh] = lg;
    atomicMaxF(&nmax[d * 2 + hh], lg);
  }
}

// Edge pass 2: a = exp(logit - max[dst]); store a; denom[dst] += a.
__global__ void k_e2(const int* __restrict__ ei, float* __restrict__ elog,
                     const float* __restrict__ nmax, float* __restrict__ den, int E) {
  int e = blockIdx.x * blockDim.x + threadIdx.x;
  if (e >= E) return;
  int d = ei[E + e];
#pragma unroll
  for (int hh = 0; hh < 2; ++hh) {
    float a = __expf(elog[(size_t)e * 2 + hh] - nmax[d * 2 + hh]);
    elog[(size_t)e * 2 + hh] = a;
    atomicAdd(&den[d * 2 + hh], a);
  }
}

// Edge pass 3: acc[dst] += xl[src] * alpha.
__global__ void k_e3(const int* __restrict__ ei, const float* __restrict__ elog,
                     const float* __restrict__ den, const float* __restrict__ xl,
                     float* __restrict__ acc, int E) {
  int e = blockIdx.x * blockDim.x + threadIdx.x;
  if (e >= E) return;
  int s = ei[e], d = ei[E + e];
  float al0 = elog[(size_t)e * 2 + 0] / (den[d * 2 + 0] + 1e-16f);
  float al1 = elog[(size_t)e * 2 + 1] / (den[d * 2 + 1] + 1e-16f);
  const float* xls = xl + (size_t)s * 16;
  float* ad = acc + (size_t)d * 16;
#pragma unroll
  for (int j = 0; j < 16; ++j) atomicAdd(&ad[j], xls[j] * (j < 8 ? al0 : al1));
}

// Layer-1 finalize: h = lrelu0.1(acc + bias) (feeds layer 2).
__global__ void k_fin1(const float* __restrict__ acc, const float* __restrict__ bias,
                       float* __restrict__ h, int N) {
  int i = blockIdx.x * blockDim.x + threadIdx.x;
  if (i >= N) return;
#pragma unroll
  for (int j = 0; j < 16; ++j) {
    float v = acc[(size_t)i * 16 + j] + bias[j];
    h[(size_t)i * 16 + j] = LRELU_MLP(v);
  }
}

// Layer-2 finalize fused with sum pool: pool[j] += acc[i,j] + bias[j]
// (LDS atomic block reduction, then 16 global atomics per block).
__global__ void k_fin2_pool(const float* __restrict__ acc,
                            const float* __restrict__ bias,
                            float* __restrict__ pool, int N) {
  __shared__ float sp[16];
  int i = blockIdx.x * blockDim.x + threadIdx.x;
  if (threadIdx.x < 16) sp[threadIdx.x] = 0.f;
  __syncthreads();
  if (i < N) {
#pragma unroll
    for (int j = 0; j < 16; ++j) atomicAdd(&sp[j], acc[(size_t)i * 16 + j] + bias[j]);
  }
  __syncthreads();
  if (threadIdx.x < 16) atomicAdd(&pool[threadIdx.x], sp[threadIdx.x]);
}

__device__ __forceinline__ float dotn(const float* __restrict__ w,
                                      const float* __restrict__ v, int n) {
  float s = 0.f;
  for (int k = 0; k < n; ++k) s += w[k] * v[k];
  return s;
}

// Tiny MLP head + softmax; negligible cost, run single-lane.
__global__ void k_tail(const float* __restrict__ pool, const float* __restrict__ meta,
                       const float* sp1w, const float* sp1b, const float* sp2w,
                       const float* sp2b, const float* mew, const float* meb,
                       const float* fc1w, const float* fc1b, const float* fc2w,
                       const float* fc2b, const float* fc3w, const float* fc3b,
                       float* __restrict__ out) {
  if (threadIdx.x != 0 || blockIdx.x != 0) return;
  float p[16];
  for (int j = 0; j < 16; ++j) p[j] = pool[j];
  float t1[64];
  for (int j = 0; j < 64; ++j) {
    float v = dotn(sp1w + j * 16, p, 16) + sp1b[j];
    t1[j] = LRELU_MLP(v);
  }
  float z[24];
  for (int j = 0; j < 16; ++j) z[j] = dotn(sp2w + j * 64, t1, 64) + sp2b[j];
  float mt[6];
  for (int j = 0; j < 6; ++j) mt[j] = meta[j];
  for (int j = 0; j < 8; ++j) {
    float v = dotn(mew + j * 6, mt, 6) + meb[j];
    z[16 + j] = LRELU_MLP(v);
  }
  float z1[24];
  for (int j = 0; j < 24; ++j) {
    float v = dotn(fc1w + j * 24, z, 24) + fc1b[j];
    z1[j] = LRELU_MLP(v);
  }
  float z2[8];
  for (int j = 0; j < 8; ++j) {
    float v = dotn(fc2w + j * 24, z1, 24) + fc2b[j];
    z2[j] = LRELU_MLP(v);
  }
  float z3[3];
  for (int j = 0; j < 3; ++j) z3[j] = dotn(fc3w + j * 8, z2, 8) + fc3b[j];
  float mx = fmaxf(z3[0], fmaxf(z3[1], z3[2]));
  float ex[3], sm = 0.f;
  for (int j = 0; j < 3; ++j) {
    ex[j] = __expf(z3[j] - mx);
    sm += ex[j];
  }
  for (int j = 0; j < 3; ++j) out[j] = ex[j] / sm;
}

extern "C" void kernel_launch(void* const* d_in, const int* in_sizes, int n_in,
                              void* d_out, int out_size, void* d_ws, size_t ws_size,
                              hipStream_t stream) {
  const float* x = (const float*)d_in[0];
  const int* ei = (const int*)d_in[1];
  const float* ea = (const float*)d_in[2];
  const float* meta = (const float*)d_in[3];
  const float* ft1_w = (const float*)d_in[4];
  const float* ft1_b = (const float*)d_in[5];
  const float* ft2_w = (const float*)d_in[6];
  const float* ft2_b = (const float*)d_in[7];
  const float* g1_wl = (const float*)d_in[8];
  const float* g1_bl = (const float*)d_in[9];
  const float* g1_wr = (const float*)d_in[10];
  const float* g1_br = (const float*)d_in[11];
  const float* g1_we = (const float*)d_in[12];
  const float* g1_att = (const float*)d_in[13];
  const float* g1_bias = (const float*)d_in[14];
  const float* g2_wl = (const float*)d_in[15];
  const float* g2_bl = (const float*)d_in[16];
  const float* g2_wr = (const float*)d_in[17];
  const float* g2_br = (const float*)d_in[18];
  const float* g2_we = (const float*)d_in[19];
  const float* g2_att = (const float*)d_in[20];
  const float* g2_bias = (const float*)d_in[21];
  const float* sp1_w = (const float*)d_in[22];
  const float* sp1_b = (const float*)d_in[23];
  const float* sp2_w = (const float*)d_in[24];
  const float* sp2_b = (const float*)d_in[25];
  const float* me_w = (const float*)d_in[26];
  const float* me_b = (const float*)d_in[27];
  const float* fc1_w = (const float*)d_in[28];
  const float* fc1_b = (const float*)d_in[29];
  const float* fc2_w = (const float*)d_in[30];
  const float* fc2_b = (const float*)d_in[31];
  const float* fc3_w = (const float*)d_in[32];
  const float* fc3_b = (const float*)d_in[33];
  float* out = (float*)d_out;

  int N = in_sizes[0] / 3;
  int E = in_sizes[2] / 2;
  int Npad = (N + 15) & ~15;
  int ntiles = Npad / 16;

  // Workspace carve (floats): h / xl / xr / acc [Npad*16] + nmax/den [N*2]
  // + edge logits [E*2] + pool [16].
  float* ws = (float*)d_ws;
  size_t o = 0;
  float* h = ws + o;    o += (size_t)Npad * 16;
  float* xl = ws + o;   o += (size_t)Npad * 16;
  float* xr = ws + o;   o += (size_t)Npad * 16;
  float* acc = ws + o;  o += (size_t)Npad * 16;
  float* nmax = ws + o; o += (size_t)N * 2;
  float* den = ws + o;  o += (size_t)N * 2;
  float* elog = ws + o; o += (size_t)E * 2;
  float* pool = ws + o;

  dim3 bN((Npad + 255) / 256), tN(256);
  dim3 bNn((N + 255) / 256);
  dim3 bE((E + 255) / 256), tE(256);
  dim3 bT((ntiles + 3) / 4), tT(128);  // 4 waves/block, 1 tile/wave

  // Node MLP (padded to 16 cols)
  k_node<<<bN, tN, 0, stream>>>(x, ft1_w, ft1_b, ft2_w, ft2_b, h, N, Npad);

  // ---- GAT layer 1 (Din = 8) ----
  k_init<<<bNn, tN, 0, stream>>>(nmax, den, acc, nullptr, N);
  k_lin<<<bT, tT, 0, stream>>>(h, g1_wl, g1_bl, g1_wr, g1_br, xl, xr, 8, ntiles);
  k_e1<<<bE, tE, 0, stream>>>(ei, ea, xl, xr, g1_we, g1_att, elog, nmax, E);
  k_e2<<<bE, tE, 0, stream>>>(ei, elog, nmax, den, E);
  k_e3<<<bE, tE, 0, stream>>>(ei, elog, den, xl, acc, E);
  k_fin1<<<bNn, tN, 0, stream>>>(acc, g1_bias, h, N);

  // ---- GAT layer 2 (Din = 16) ----
  k_init<<<bNn, tN, 0, stream>>>(nmax, den, acc, pool, N);
  k_lin<<<bT, tT, 0, stream>>>(h, g2_wl, g2_bl, g2_wr, g2_br, xl, xr, 16, ntiles);
  k_e1<<<bE, tE, 0, stream>>>(ei, ea, xl, xr, g2_we, g2_att, elog, nmax, E);
  k_e2<<<bE, tE, 0, stream>>>(ei, elog, nmax, den, E);
  k_e3<<<bE, tE, 0, stream>>>(ei, elog, den, xl, acc, E);
  k_fin2_pool<<<bNn, tN, 0, stream>>>(acc, g2_bias, pool, N);

  // ---- head MLP + softmax ----
  k_tail<<<1, 32, 0, stream>>>(pool, meta, sp1_w, sp1_b, sp2_w, sp2_b, me_w, me_b,
                               fc1_w, fc1_b, fc2_w, fc2_b, fc3_w, fc3_b, out);
}